// GCNEncoder_82188494176916
// MI455X (gfx1250) — compile-verified
//
#include <hip/hip_runtime.h>

#define N_NODES 50000
#define N_EDGES 800000
#define IN_C 128
#define HID_C 256
#define OUT_C 128

typedef __attribute__((ext_vector_type(2))) float v2f;
typedef __attribute__((ext_vector_type(8))) float v8f;

// ---------------------------------------------------------------------------
// Degree / normalization
// ---------------------------------------------------------------------------
__global__ void init_deg_kernel(float* __restrict__ deg, int n) {
    int i = blockIdx.x * blockDim.x + threadIdx.x;
    if (i < n) deg[i] = 1.0f;  // self-loop contributes 1
}

__global__ void count_deg_kernel(const int* __restrict__ dst, float* __restrict__ deg, int nE) {
    int e = blockIdx.x * blockDim.x + threadIdx.x;
    if (e < nE) atomicAdd(&deg[dst[e]], 1.0f);
}

__global__ void dinv_kernel(float* __restrict__ deg, int n) {
    int i = blockIdx.x * blockDim.x + threadIdx.x;
    if (i < n) deg[i] = rsqrtf(deg[i]);  // deg >= 1 always (self-loops)
}

// ---------------------------------------------------------------------------
// Self-loop init: T[i,c] = (bias ? bias[c] : 0) + X[i,c] * dinv[i]^2
// 128 features per node, one wave (32 lanes x float4) per node.
// ---------------------------------------------------------------------------
__global__ void self_init_kernel(const float* __restrict__ X, const float* __restrict__ dinv,
                                 const float* __restrict__ bias, float* __restrict__ T, int n) {
    int gid  = blockIdx.x * blockDim.x + threadIdx.x;
    int node = gid >> 5;
    int lane = gid & 31;
    if (node >= n) return;
    float w = dinv[node];
    w = w * w;
    float4 v = ((const float4*)(X + (size_t)node * 128))[lane];
    float4 o;
    o.x = v.x * w; o.y = v.y * w; o.z = v.z * w; o.w = v.w * w;
    if (bias) {
        float4 bb = ((const float4*)bias)[lane];
        o.x += bb.x; o.y += bb.y; o.z += bb.z; o.w += bb.w;
    }
    ((float4*)(T + (size_t)node * 128))[lane] = o;
}

// ---------------------------------------------------------------------------
// Edge scatter: Out[dst] += X[src] * dinv[src]*dinv[dst]   (128 features)
// One wave per edge; each lane handles 4 consecutive features.
// ---------------------------------------------------------------------------
__global__ void edge_agg_kernel(const int* __restrict__ src, const int* __restrict__ dst,
                                const float* __restrict__ dinv, const float* __restrict__ X,
                                float* __restrict__ Out, int nE) {
    int gid  = blockIdx.x * blockDim.x + threadIdx.x;
    int e    = gid >> 5;
    int lane = gid & 31;
    if (e >= nE) return;
    int s = src[e];
    int d = dst[e];
    float w = dinv[s] * dinv[d];
    float4 v = ((const float4*)(X + (size_t)s * 128))[lane];
    float* o = Out + (size_t)d * 128 + lane * 4;
    atomicAdd(o + 0, v.x * w);
    atomicAdd(o + 1, v.y * w);
    atomicAdd(o + 2, v.z * w);
    atomicAdd(o + 3, v.w * w);
}

// ---------------------------------------------------------------------------
// WMMA f32 GEMM: C[M,NC] = A[M,K] @ W[K,NC] (+bias) (opt relu), M = N_NODES.
//
// Block = 256 threads (8 waves). Each block owns one 64-column weight panel
// (gn) and 8 consecutive 16-row M-tiles (one per wave). The weight panel is
// staged into LDS in 128-row K-chunks via GLOBAL_LOAD_ASYNC_TO_LDS_B128
// (ASYNCcnt), then each wave runs V_WMMA_F32_16X16X4_F32 with:
//   A fragment (16x4): lanes 0-15 -> M=lane, K={k,k+1}; lanes 16-31 K={k+2,k+3}
//   B fragment (4x16): mirrored, read from LDS (rows padded to 80 floats so
//     the two lane-halves (rows k vs k+2) sit 32 banks apart -> conflict-free)
//   C/D (16x16, 8 VGPR): VGPR r: lanes 0-15 (M=r,N=lane); lanes16-31 (M=r+8)
// ---------------------------------------------------------------------------
template <int K, int NC>
__launch_bounds__(256)
__global__ void gemm_wmma_kernel(const float* __restrict__ A, const float* __restrict__ W,
                                 const float* __restrict__ bias, float* __restrict__ C,
                                 int doRelu) {
    constexpr int NT      = 4;             // 4 x 16 = 64 output columns per wave
    constexpr int GROUPSN = NC / 64;
    constexpr int KC      = 128;           // K-chunk resident in LDS
    constexpr int LDSTR   = 80;            // padded LDS row stride (floats)
    constexpr int TILES_M = N_NODES / 16;  // 3125 (exact)

    extern __shared__ float ldsW[];        // KC * LDSTR * 4 = 40960 bytes

    const int gn   = blockIdx.x % GROUPSN;
    const int mb   = blockIdx.x / GROUPSN;
    const int tid  = threadIdx.x;
    const int wave = tid >> 5;
    const int lane = tid & 31;
    const int tm   = mb * 8 + wave;
    const bool valid = (tm < TILES_M);     // wave-uniform -> EXEC all-ones for WMMA

    const int half = lane >> 4;            // 0: lanes 0-15, 1: lanes 16-31
    const int l15  = lane & 15;

    // Clamp row for invalid waves so no OOB address is ever formed.
    const int arow = (valid ? tm : 0) * 16 + l15;
    const float* Ap = A + (size_t)arow * K + 2 * half;

    v8f acc[NT] = {};

    for (int kc = 0; kc < K; kc += KC) {
        if (kc) __syncthreads();  // previous chunk fully consumed before overwrite

        // --- cooperative async stage of W[kc..kc+KC) x [gn*64..gn*64+64) ---
#pragma unroll
        for (int it = 0; it < (KC * 16) / 256; ++it) {  // 8 iters: 2048 float4s
            int idx = it * 256 + tid;
            int row = idx >> 4;           // 0..127 within chunk
            int c4  = idx & 15;           // float4 column within panel
            unsigned ldsOff = (unsigned)((row * LDSTR + c4 * 4) * 4);
            const float* gp = W + (size_t)(kc + row) * NC + gn * 64 + c4 * 4;
            asm volatile("global_load_async_to_lds_b128 %0, %1, off"
                         :: "v"(ldsOff), "v"(gp)
                         : "memory");
        }
        asm volatile("s_wait_asynccnt 0x0" ::: "memory");
        __syncthreads();

        if (valid) {
            const float* Bbase = ldsW + half * 2 * LDSTR + l15;
#pragma unroll 4
            for (int kk = 0; kk < KC; kk += 4) {
                v2f a;
                a[0] = Ap[kc + kk];
                a[1] = Ap[kc + kk + 1];
                const float* bp = Bbase + kk * LDSTR;
#pragma unroll
                for (int t = 0; t < NT; ++t) {
                    v2f b;
                    b[0] = bp[16 * t];
                    b[1] = bp[LDSTR + 16 * t];
                    acc[t] = __builtin_amdgcn_wmma_f32_16x16x4_f32(
                        /*neg_a=*/false, a, /*neg_b=*/false, b,
                        /*c_mod=*/(short)0, acc[t],
                        /*reuse_a=*/false, /*reuse_b=*/false);
                }
            }
        }
    }

    if (valid) {
#pragma unroll
        for (int t = 0; t < NT; ++t) {
            const int col = gn * 64 + 16 * t + l15;
            const float bv = bias ? bias[col] : 0.0f;
#pragma unroll
            for (int r = 0; r < 8; ++r) {
                const int row = tm * 16 + r + 8 * half;
                float v = acc[t][r] + bv;
                if (doRelu) v = fmaxf(v, 0.0f);
                C[(size_t)row * NC + col] = v;
            }
        }
    }
}

// ---------------------------------------------------------------------------
// Launch
// ---------------------------------------------------------------------------
extern "C" void kernel_launch(void* const* d_in, const int* in_sizes, int n_in,
                              void* d_out, int out_size, void* d_ws, size_t ws_size,
                              hipStream_t stream) {
    const float* x  = (const float*)d_in[0];          // [N, 128]
    const int*   ei = (const int*)d_in[1];            // [2, E]
    const float* W1 = (const float*)d_in[2];          // [128, 256]
    const float* b1 = (const float*)d_in[3];          // [256]
    const float* W2 = (const float*)d_in[4];          // [256, 128]
    const float* b2 = (const float*)d_in[5];          // [128]
    float* out = (float*)d_out;                       // [N, 128]

    const int* src = ei;
    const int* dst = ei + N_EDGES;

    // Workspace layout (256B aligned)
    char* ws = (char*)d_ws;
    constexpr size_t DINV_BYTES = (((size_t)N_NODES * 4) + 255) & ~(size_t)255;   // 200192
    constexpr size_t AGG1_BYTES = (size_t)N_NODES * IN_C * 4;                     // 25.6 MB
    constexpr size_t H_BYTES    = (size_t)N_NODES * HID_C * 4;                    // 51.2 MB
    float* dinv = (float*)(ws);
    float* agg1 = (float*)(ws + DINV_BYTES);
    float* h    = (float*)(ws + DINV_BYTES + AGG1_BYTES);
    float* hw2  = (float*)(ws + DINV_BYTES + AGG1_BYTES + H_BYTES);

    constexpr int TPB = 256;  // 8 waves of 32
    constexpr int NODE_BLKS  = (N_NODES + TPB - 1) / TPB;
    constexpr int EDGE_BLKS  = (N_EDGES + TPB - 1) / TPB;
    constexpr int NWAVE_BLKS = ((size_t)N_NODES * 32 + TPB - 1) / TPB;   // wave per node
    constexpr int EWAVE_BLKS = ((size_t)N_EDGES * 32 + TPB - 1) / TPB;   // wave per edge

    constexpr int TILES_M  = N_NODES / 16;            // 3125
    constexpr int MB       = (TILES_M + 7) / 8;       // 391 blocks along M
    constexpr int G1_BLKS  = MB * (HID_C / 64);       // 1564
    constexpr int G2_BLKS  = MB * (OUT_C / 64);       // 782
    constexpr size_t GEMM_LDS = 128 * 80 * 4;         // 40960 B per block

    // 1) Normalization: deg -> dinv
    init_deg_kernel<<<NODE_BLKS, TPB, 0, stream>>>(dinv, N_NODES);
    count_deg_kernel<<<EDGE_BLKS, TPB, 0, stream>>>(dst, dinv, N_EDGES);
    dinv_kernel<<<NODE_BLKS, TPB, 0, stream>>>(dinv, N_NODES);

    // 2) Layer 1: aggregate x first (128 feats), then GEMM
    self_init_kernel<<<NWAVE_BLKS, TPB, 0, stream>>>(x, dinv, nullptr, agg1, N_NODES);
    edge_agg_kernel<<<EWAVE_BLKS, TPB, 0, stream>>>(src, dst, dinv, x, agg1, N_EDGES);
    //    h = relu(agg1 @ W1 + b1)
    gemm_wmma_kernel<IN_C, HID_C><<<G1_BLKS, TPB, GEMM_LDS, stream>>>(agg1, W1, b1, h, /*relu=*/1);

    // 3) Layer 2: GEMM first, then aggregate the 128-feat result
    //    hw2 = h @ W2
    gemm_wmma_kernel<HID_C, OUT_C><<<G2_BLKS, TPB, GEMM_LDS, stream>>>(h, W2, nullptr, hw2, /*relu=*/0);
    //    out = b2 + self + scatter(hw2)
    self_init_kernel<<<NWAVE_BLKS, TPB, 0, stream>>>(hw2, dinv, b2, out, N_NODES);
    edge_agg_kernel<<<EWAVE_BLKS, TPB, 0, stream>>>(src, dst, dinv, hw2, out, N_EDGES);
}